// unit_gcn_54563264528966
// MI455X (gfx1250) — compile-verified
//
#include <hip/hip_runtime.h>
#include <math.h>

// ---------------------------------------------------------------------------
// unit_gcn forward for MI455X (gfx1250, wave32, WMMA)
// Shapes: N=64, C=V=25, T=512, O=64, NA=3, R=12, NT=N*T=32768
// ---------------------------------------------------------------------------

typedef float v2f __attribute__((ext_vector_type(2)));
typedef float v4f __attribute__((ext_vector_type(4)));
typedef float v8f __attribute__((ext_vector_type(8)));

#define NW 4          // waves per block in k1
#define NT_TOT 32768
#define T_DIM 512
#define V_DIM 25
#define O_DIM 64
#define KPAD 52       // 50 real K (xa1:25 + xa0:25) padded to multiple of 4

static __device__ __forceinline__ v8f wmma_f32(v2f a, v2f b, v8f c) {
    // V_WMMA_F32_16X16X4_F32 : D = A(16x4) * B(4x16) + C(16x16), fp32, wave32
    return __builtin_amdgcn_wmma_f32_16x16x4_f32(
        /*neg_a=*/false, a, /*neg_b=*/false, b,
        /*c_mod=*/(short)0, c, /*reuse_a=*/false, /*reuse_b=*/false);
}

__global__ __launch_bounds__(128) void unit_gcn_k1(
    const float* __restrict__ x,      // (64,25,512,25)
    const float* __restrict__ A,      // (3,25,25)
    const float* __restrict__ se_w1,  // (12,25)
    const float* __restrict__ se_b1,  // (12)
    const float* __restrict__ se_w2,  // (25,12)
    const float* __restrict__ se_b2,  // (25)
    const float* __restrict__ conv_w, // (3,64,25)
    const float* __restrict__ conv_b, // (3,64)
    float* __restrict__ out,          // (64,64,512,25) pre-BN
    float* __restrict__ ws)           // [0:64) sum, [64:128) sumsq
{
    // ---- block-shared, read-mostly ----
    __shared__ float WcatS[O_DIM][KPAD];   // [o][k]: k<25 -> W0, 25..49 -> W1+W2, 50/51 -> 0
    __shared__ float A1S[V_DIM][V_DIM];
    __shared__ float A0dS[V_DIM];
    __shared__ float bsumS[O_DIM];
    __shared__ float bnSum[O_DIM];
    __shared__ float bnSq[O_DIM];
    // ---- per-wave scratch ----
    __shared__ float XtS[NW][V_DIM][V_DIM];   // Xt[c][v]
    __shared__ float xxxS[NW][28][V_DIM];     // xxx[v][c]; rows 25..27 zero (K pad)
    __shared__ float XAs[NW][KPAD][V_DIM];    // rows 0..24 xa1, 25..49 xa0, 50..51 zero
    __shared__ float pooledS[NW][V_DIM];
    __shared__ float hS[NW][12];
    __shared__ float wS[NW][V_DIM];
    __shared__ float dS[NW][V_DIM];

    const int tid  = threadIdx.x;
    const int wv   = tid >> 5;
    const int lane = tid & 31;
    const int half = lane >> 4;   // 0: lanes 0-15, 1: lanes 16-31
    const int lm   = lane & 15;

    // ---- shared setup ----
    for (int i = tid; i < O_DIM * KPAD; i += 128) {
        int o = i / KPAD, k = i % KPAD;
        float v = 0.f;
        if (k < 25)       v = conv_w[o * 25 + k];
        else if (k < 50)  { int c = k - 25; v = conv_w[1600 + o * 25 + c] + conv_w[3200 + o * 25 + c]; }
        WcatS[o][k] = v;
    }
    for (int i = tid; i < 625; i += 128) A1S[i / 25][i % 25] = A[625 + i];
    if (tid < V_DIM) A0dS[tid] = A[tid * 25 + tid];
    if (tid < O_DIM) {
        bsumS[tid] = conv_b[tid] + conv_b[64 + tid] + conv_b[128 + tid];
        bnSum[tid] = 0.f;
        bnSq[tid]  = 0.f;
    }

    // ---- one (n,t) per wave ----
    const int nt = blockIdx.x * NW + wv;
    const int n  = nt >> 9;          // nt / 512
    const int t  = nt & 511;
    const float* xb = x + ((size_t)n * 25 * T_DIM + t) * V_DIM;   // x[n][c][t][v] = xb[c*12800 + v]

    for (int i = lane; i < 625; i += 32) {
        int c = i / 25, v = i % 25;
        XtS[wv][c][v] = xb[(size_t)c * (T_DIM * V_DIM) + v];
    }
    // zero K-pad regions (must be true zeros: 0 * garbage-NaN would poison WMMA)
    for (int i = lane; i < 75; i += 32) xxxS[wv][25 + i / 25][i % 25] = 0.f;
    for (int i = lane; i < 50; i += 32) XAs[wv][50 + i / 25][i % 25] = 0.f;
    __syncthreads();

    // pooled[v] = mean_c Xt[c][v]
    if (lane < V_DIM) {
        float s = 0.f;
        for (int c = 0; c < 25; ++c) s += XtS[wv][c][lane];
        pooledS[wv][lane] = s * (1.f / 25.f);
    }
    __syncthreads();
    // h = relu(pooled @ w1^T + b1)
    if (lane < 12) {
        float s = se_b1[lane];
        for (int v = 0; v < 25; ++v) s += pooledS[wv][v] * se_w1[lane * 25 + v];
        hS[wv][lane] = fmaxf(s, 0.f);
    }
    __syncthreads();
    // w = sigmoid(h @ w2^T + b2)
    if (lane < V_DIM) {
        float s = se_b2[lane];
        for (int r = 0; r < 12; ++r) s += hS[wv][r] * se_w2[lane * 12 + r];
        wS[wv][lane] = 1.f / (1.f + expf(-s));
    }
    __syncthreads();
    // d[i] = A0[i,i] * (sum_v Xt[i,v]*w[v] - Xt[i,i]*w[i])
    if (lane < V_DIM) {
        float s = 0.f;
        for (int v = 0; v < 25; ++v) s += XtS[wv][lane][v] * wS[wv][v];
        s -= XtS[wv][lane][lane] * wS[wv][lane];
        dS[wv][lane] = A0dS[lane] * s;
    }
    __syncthreads();
    // xxx[v][c] = Xt[c][v]*w[v]*A1[v][c]; XA rows 25..49 = xa0[c][v] = Xt[c][v]*d[v]
    for (int i = lane; i < 625; i += 32) {
        int a = i / 25, b = i % 25;
        xxxS[wv][a][b]     = XtS[wv][b][a] * wS[wv][a] * A1S[a][b];
        XAs[wv][25 + a][b] = XtS[wv][a][b] * dS[wv][b];
    }
    __syncthreads();

    // ---- stage 1 (WMMA fp32): xa1(25x25) = Xt(25x25) @ xxx(25x25) -> XA rows 0..24
    // M/N out-of-range lanes are CLAMPED (their D rows/cols are discarded; clamped
    // values are finite so no NaN can leak into kept elements). K-pad rows of B are
    // true zeros; A's k is clamped so 0*finite = 0.
    for (int mm = 0; mm < 32; mm += 16) {
        const int mA = min(mm + lm, 24);
        for (int nn = 0; nn < 32; nn += 16) {
            const int bn = min(nn + lm, 24);
            v8f acc = {0.f, 0.f, 0.f, 0.f, 0.f, 0.f, 0.f, 0.f};
            for (int kk = 0; kk < 28; kk += 4) {
                v2f a, b;
#pragma unroll
                for (int j = 0; j < 2; ++j) {
                    int k = kk + j + 2 * half;
                    a[j] = XtS[wv][mA][min(k, 24)];
                    b[j] = xxxS[wv][k][bn];
                }
                acc = wmma_f32(a, b, acc);
            }
#pragma unroll
            for (int g = 0; g < 8; ++g) {
                int r = mm + 8 * half + g, cc = nn + lm;
                if (r < 25 && cc < 25) XAs[wv][r][cc] = acc[g];
            }
        }
    }
    __syncthreads();

    // ---- stage 2 (WMMA fp32): Y(64x25) = Wcat(64x52) @ XA(52x25) + bsum
    float* yout = out + ((size_t)n * O_DIM * T_DIM + t) * V_DIM;  // yout[o*12800 + v]
    for (int mm = 0; mm < O_DIM; mm += 16) {
        const int mA = mm + lm;                      // always < 64, no clamp needed
        for (int nn = 0; nn < 32; nn += 16) {
            const int bn = min(nn + lm, 24);
            v8f acc;
#pragma unroll
            for (int g = 0; g < 8; ++g) acc[g] = bsumS[mm + 8 * half + g];
            for (int kk = 0; kk < KPAD; kk += 4) {
                v2f a, b;
#pragma unroll
                for (int j = 0; j < 2; ++j) {
                    int k = kk + j + 2 * half;
                    a[j] = WcatS[mA][k];
                    b[j] = XAs[wv][k][bn];
                }
                acc = wmma_f32(a, b, acc);
            }
#pragma unroll
            for (int g = 0; g < 8; ++g) {
                int o = mm + 8 * half + g, v = nn + lm;
                float val = acc[g];
                bool ok = (v < 25);
                // all 16 lanes of a half share the same o -> shfl-reduce over lm
                float s = ok ? val : 0.f;
                float q = s * s;
#pragma unroll
                for (int off = 1; off < 16; off <<= 1) {
                    s += __shfl_xor(s, off, 32);
                    q += __shfl_xor(q, off, 32);
                }
                if (ok) __builtin_nontemporal_store(val, &yout[(size_t)o * (T_DIM * V_DIM) + v]);
                if (lm == 0) {
                    atomicAdd(&bnSum[o], s);   // ds_add_f32, 1 per 16 lanes
                    atomicAdd(&bnSq[o],  q);
                }
            }
        }
    }
    __syncthreads();

    // ---- flush BN partials ----
    if (tid < O_DIM) {
        unsafeAtomicAdd(&ws[tid],      bnSum[tid]);   // global_atomic_add_f32
        unsafeAtomicAdd(&ws[64 + tid], bnSq[tid]);
    }
}

__global__ void unit_gcn_k2(const float* __restrict__ gamma,
                            const float* __restrict__ beta,
                            float* __restrict__ ws)
{
    int o = threadIdx.x;
    if (o < O_DIM) {
        const float cnt = 819200.0f;  // N*T*V = 64*512*25
        float mean = ws[o] / cnt;
        float var  = ws[64 + o] / cnt - mean * mean;
        float sc   = gamma[o] * rsqrtf(var + 1e-5f);
        ws[128 + o] = sc;
        ws[192 + o] = beta[o] - mean * sc;
    }
}

__global__ __launch_bounds__(256) void unit_gcn_k3(
    float* __restrict__ out, const float* __restrict__ ws,
    const float* __restrict__ mask)
{
    __shared__ float s[128];
    if (threadIdx.x < 128) s[threadIdx.x] = ws[128 + threadIdx.x];
    __syncthreads();
    // y: 52,428,800 floats = 13,107,200 float4; o is constant within a float4
    const size_t y4 = 13107200ull;
    const size_t stride = (size_t)gridDim.x * blockDim.x;
    v4f* out4 = (v4f*)out;
    for (size_t i = (size_t)blockIdx.x * blockDim.x + threadIdx.x; i < y4; i += stride) {
        int o = (int)((i / 3200) & 63);          // (i*4 / 12800) % 64
        v4f v = out4[i];
        v4f r = v * s[o] + s[64 + o];
#pragma unroll
        for (int j = 0; j < 4; ++j) r[j] = fmaxf(r[j], 0.f);
        __builtin_nontemporal_store(r, out4 + i);
    }
    size_t gid = (size_t)blockIdx.x * blockDim.x + threadIdx.x;
    if (gid < 625) out[(size_t)52428800 + gid] = mask[gid];
}

extern "C" void kernel_launch(void* const* d_in, const int* in_sizes, int n_in,
                              void* d_out, int out_size, void* d_ws, size_t ws_size,
                              hipStream_t stream) {
    const float* x      = (const float*)d_in[0];
    const float* A      = (const float*)d_in[1];
    const float* se_w1  = (const float*)d_in[2];
    const float* se_b1  = (const float*)d_in[3];
    const float* se_w2  = (const float*)d_in[4];
    const float* se_b2  = (const float*)d_in[5];
    const float* conv_w = (const float*)d_in[6];
    const float* conv_b = (const float*)d_in[7];
    const float* gamma  = (const float*)d_in[8];
    const float* beta   = (const float*)d_in[9];
    const float* mask   = (const float*)d_in[10];
    float* out = (float*)d_out;
    float* ws  = (float*)d_ws;

    // zero BN-stat accumulators (graph-capture safe)
    hipMemsetAsync(ws, 0, 128 * sizeof(float), stream);

    unit_gcn_k1<<<NT_TOT / NW, 128, 0, stream>>>(x, A, se_w1, se_b1, se_w2, se_b2,
                                                 conv_w, conv_b, out, ws);
    unit_gcn_k2<<<1, 64, 0, stream>>>(gamma, beta, ws);
    unit_gcn_k3<<<4096, 256, 0, stream>>>(out, ws, mask);
}